// PoolEdgesToVertices_1529008357760
// MI455X (gfx1250) — compile-verified
//
#include <hip/hip_runtime.h>
#include <hip/hip_bf16.h>

typedef float v2f __attribute__((ext_vector_type(2)));
typedef float v8f __attribute__((ext_vector_type(8)));

#define V_N 4096
#define E_N 32768
#define D_N 32

#define WPB   8                 // waves per block (wave32)
#define TPB   (WPB * 32)        // 256 threads
#define KSPAN (E_N / WPB)       // 4096 K-elements per wave
#define KSUP  32                // K per superstep (one 128B line per row)

// C[V,D] = conEd[V,E] @ e_in[E,D] ; out_v = v_in + C ; fused: out_conEd = conEd.
// One block = one 16-row M-tile; its 8 waves split K and partial 16x32
// accumulators are reduced through LDS by wave 0.
__global__ __launch_bounds__(TPB) void pool_edges_wmma_f32(
    const float* __restrict__ v_in,
    const float* __restrict__ e_in,
    const float* __restrict__ conEd,
    float* __restrict__ out_v,
    float* __restrict__ out_conEd)
{
    __shared__ float red[WPB * 32 * 16];   // 16 KB: per-wave 16x32 partials

    const int tid  = threadIdx.x;
    const int w    = tid >> 5;
    const int lane = tid & 31;
    const int half = lane >> 4;     // 0: lanes 0-15, 1: lanes 16-31
    const int mrow = lane & 15;

    const int m0 = blockIdx.x * 16;               // 256 blocks cover V=4096
    const size_t rowBase = (size_t)(m0 + mrow) * E_N;

    v8f acc0 = {};   // N = 0..15
    v8f acc1 = {};   // N = 16..31

    const int k_begin = w * KSPAN;
    const int k_end   = k_begin + KSPAN;

    for (int k0 = k_begin; k0 < k_end; k0 += KSUP) {
        // Stream prefetch, 2KB (16 supersteps) ahead, near-RT policy so it
        // lands in the WGP cache (global_prefetch_b8). One per 128B line/row.
        __builtin_prefetch(conEd + rowBase + k0 + 2 * half + 512, 0, 3);

        #pragma unroll
        for (int kk = 0; kk < KSUP; kk += 4) {
            const int kb = k0 + kk + 2 * half;

            // A fragment: 16x4 f32. Lane holds conEd[m0+mrow][kb], [kb+1].
            v2f a = *(const v2f*)(conEd + rowBase + kb);
            // Fused pass-through copy of conEd (single HBM read, not two).
            *(v2f*)(out_conEd + rowBase + kb) = a;

            // B fragments: 4x16 f32 per N-half. Lane: e_in[kb][n], e_in[kb+1][n].
            v2f b0, b1;
            b0.x = e_in[(size_t)kb       * D_N + mrow];
            b0.y = e_in[(size_t)(kb + 1) * D_N + mrow];
            b1.x = e_in[(size_t)kb       * D_N + 16 + mrow];
            b1.y = e_in[(size_t)(kb + 1) * D_N + 16 + mrow];

            acc0 = __builtin_amdgcn_wmma_f32_16x16x4_f32(
                false, a, false, b0, (short)0, acc0, false, false);
            acc1 = __builtin_amdgcn_wmma_f32_16x16x4_f32(
                false, a, false, b1, (short)0, acc1, false, false);
        }
    }

    // ---- cross-wave K reduction through LDS ----
    float* my = red + ((w << 5) + lane) * 16;
    #pragma unroll
    for (int j = 0; j < 8; ++j) { my[j] = acc0[j]; my[8 + j] = acc1[j]; }
    __syncthreads();

    if (w == 0) {
        #pragma unroll
        for (int ww = 1; ww < WPB; ++ww) {
            const float* other = red + ((ww << 5) + lane) * 16;
            #pragma unroll
            for (int j = 0; j < 8; ++j) {
                acc0[j] += other[j];
                acc1[j] += other[8 + j];
            }
        }
        // C/D layout: VGPR j -> M = m0 + j + 8*half, N = lane&15 (+ n0).
        #pragma unroll
        for (int j = 0; j < 8; ++j) {
            const int mm = m0 + j + 8 * half;
            const size_t b = (size_t)mm * D_N + mrow;
            out_v[b]      = v_in[b]      + acc0[j];
            out_v[b + 16] = v_in[b + 16] + acc1[j];
        }
    }
}

extern "C" void kernel_launch(void* const* d_in, const int* in_sizes, int n_in,
                              void* d_out, int out_size, void* d_ws, size_t ws_size,
                              hipStream_t stream) {
    const float* v_in  = (const float*)d_in[0];   // [V,D]
    const float* e_in  = (const float*)d_in[1];   // [E,D]
    const float* u_in  = (const float*)d_in[2];   // [1,D]
    const float* adj   = (const float*)d_in[3];   // [V,V]
    const float* conEd = (const float*)d_in[4];   // [V,E]

    float* out = (float*)d_out;
    const size_t n_v   = (size_t)V_N * D_N;       //   131072
    const size_t n_e   = (size_t)E_N * D_N;       //  1048576
    const size_t n_u   = (size_t)D_N;             //       32
    const size_t n_adj = (size_t)V_N * V_N;       // 16777216

    float* out_v     = out;
    float* out_e     = out_v + n_v;
    float* out_u     = out_e + n_e;
    float* out_adj   = out_u + n_u;
    float* out_conEd = out_adj + n_adj;

    // Pass-through outputs: pure streaming copies, graph-capture-safe async D2D.
    hipMemcpyAsync(out_e,   e_in, n_e   * sizeof(float), hipMemcpyDeviceToDevice, stream);
    hipMemcpyAsync(out_u,   u_in, n_u   * sizeof(float), hipMemcpyDeviceToDevice, stream);
    hipMemcpyAsync(out_adj, adj,  n_adj * sizeof(float), hipMemcpyDeviceToDevice, stream);

    // Fused GEMM + bias-add + conEd copy: 256 M-tiles, 8 K-split waves each.
    dim3 grid(V_N / 16);   // 256
    dim3 block(TPB);       // 256 = 8 wave32
    pool_edges_wmma_f32<<<grid, block, 0, stream>>>(v_in, e_in, conEd, out_v, out_conEd);
}